// MeanAggregator_43946105372917
// MI455X (gfx1250) — compile-verified
//
#include <hip/hip_runtime.h>
#include <stdint.h>

// MeanAggregator: out[b,:] = mean_k features[neigh_idx[b,k],:]
// U=1e6 rows x D=128 f32, B=1e5, K=10.
// Pure bandwidth problem (~567 MB @ 23.3 TB/s ~= 24 us floor, 0.23 FLOP/byte).
// Strategy: one wave32 per output node; each neighbor row (512 B) is fetched
// with a single ASYNCcnt-tracked global_load_async_to_lds_b128 (32 lanes x 16 B),
// all K=10 gathers in flight per wave, drained once with s_wait_asynccnt 0,
// then reduced out of LDS with ds_load_b128.

#define WAVES_PER_BLOCK 8
#define KNBR 10
#define DIM 128

__global__ __launch_bounds__(256) void MeanAggregator_43946105372917_kernel(
    const float* __restrict__ features,
    const int*   __restrict__ neigh_idx,
    float*       __restrict__ out,
    int B)
{
    // 8 waves * 10 rows * 128 floats = 40 KB LDS per workgroup (<< 320 KB/WGP)
    __shared__ float smem[WAVES_PER_BLOCK * KNBR * DIM];

    const int lane = threadIdx.x & 31;
    const int wave = threadIdx.x >> 5;

    // Uniform per wave -> force to SGPR so index loads are scalar (s_load).
    int node = __builtin_amdgcn_readfirstlane((int)(blockIdx.x * WAVES_PER_BLOCK + wave));
    if (node >= B) return;

    const int col = lane * 4;                      // 4 consecutive floats per lane

    // ---- Stage 1: issue K async gathers (global -> LDS), ASYNCcnt-tracked ----
#pragma unroll
    for (int k = 0; k < KNBR; ++k) {
        const int idx = neigh_idx[node * KNBR + k];                 // scalar load
        const float* src = features + (size_t)(unsigned)idx * DIM + col;
        // Generic shared pointer: low 32 bits are the raw LDS byte offset
        uint32_t lds_off =
            (uint32_t)(uintptr_t)&smem[(wave * KNBR + k) * DIM + col];
        asm volatile("global_load_async_to_lds_b128 %0, %1, off"
                     :: "v"(lds_off), "v"(src)
                     : "memory");
    }

    // ---- Stage 2: drain async copies, reduce from LDS ----
    asm volatile("s_wait_asynccnt 0" ::: "memory");

    float4 acc = make_float4(0.f, 0.f, 0.f, 0.f);
#pragma unroll
    for (int k = 0; k < KNBR; ++k) {
        const float4 v =
            *(const float4*)&smem[(wave * KNBR + k) * DIM + col];   // ds_load_b128
        acc.x += v.x; acc.y += v.y; acc.z += v.z; acc.w += v.w;
    }
    const float s = 1.0f / (float)KNBR;
    acc.x *= s; acc.y *= s; acc.z *= s; acc.w *= s;

    *(float4*)(out + (size_t)node * DIM + col) = acc;               // global_store_b128
}

extern "C" void kernel_launch(void* const* d_in, const int* in_sizes, int n_in,
                              void* d_out, int out_size, void* d_ws, size_t ws_size,
                              hipStream_t stream) {
    const float* features  = (const float*)d_in[0];
    const int*   neigh_idx = (const int*)d_in[1];
    float*       out       = (float*)d_out;

    const int B = in_sizes[1] / KNBR;   // neigh_idx holds B*K indices

    dim3 block(256);                    // 8 wave32 per block
    dim3 grid((B + WAVES_PER_BLOCK - 1) / WAVES_PER_BLOCK);
    hipLaunchKernelGGL(MeanAggregator_43946105372917_kernel, grid, block, 0, stream,
                       features, neigh_idx, out, B);
}